// ComposeTransform_10728828305752
// MI455X (gfx1250) — compile-verified
//
#include <hip/hip_runtime.h>
#include <stdint.h>

// ComposeTransform: out(x) = trilerp(d1, x + d2(x)) + d2(x)
// Shapes: (B=2, D=160, H=192, W=224, C=3), float32, channel order (z,y,x).
//
// Memory-bound gather kernel: one thread per voxel, one block per W-row.
// WMMA intentionally not used: data-dependent 8-corner gather is not a dense
// contraction; the roofline is 23.3 TB/s HBM (~21 us floor), not matrix TOPS.
//
// v3: byte offsets computed directly in 32-bit on char* bases so the backend
// can emit saddr+voffset global loads (scalar 64-bit base + zext 32-bit
// vector offset) instead of per-lane 64-bit address arithmetic.

namespace {
constexpr uint32_t Wd = 224;
constexpr uint32_t Hd = 192;
constexpr uint32_t Dd = 160;
constexpr uint32_t Bd = 2;

__device__ __forceinline__ const float* byte_off(const float* base, uint32_t bytes) {
    return (const float*)((const char*)base + bytes);
}
__device__ __forceinline__ float* byte_off(float* base, uint32_t bytes) {
    return (float*)((char*)base + bytes);
}
}

__global__ __launch_bounds__(Wd) void compose_disp_kernel(
    const float* __restrict__ d1,
    const float* __restrict__ d2,
    float* __restrict__ out)
{
    const uint32_t x = threadIdx.x;   // 0..223  (fastest dim -> coalesced)
    const uint32_t y = blockIdx.x;    // 0..191
    const uint32_t z = blockIdx.y;    // 0..159
    const uint32_t b = blockIdx.z;    // 0..1

    const uint32_t vox = ((b * Dd + z) * Hd + y) * Wd + x;
    const uint32_t voxB = vox * 12u;             // byte offset, < 2^31

    // --- streamed d2 read (single use -> non-temporal, keep L2 for d1) ---
    const float* p2 = byte_off(d2, voxB);
    const float d2z = __builtin_nontemporal_load(p2 + 0);
    const float d2y = __builtin_nontemporal_load(p2 + 1);
    const float d2x = __builtin_nontemporal_load(p2 + 2);

    // --- sample location with edge clamp (matches reference) ---
    const float lz = fminf(fmaxf((float)z + d2z, 0.0f), (float)(Dd - 1));
    const float ly = fminf(fmaxf((float)y + d2y, 0.0f), (float)(Hd - 1));
    const float lx = fminf(fmaxf((float)x + d2x, 0.0f), (float)(Wd - 1));

    const float fz = floorf(lz), fy = floorf(ly), fx = floorf(lx);
    const float wz1 = lz - fz, wy1 = ly - fy, wx1 = lx - fx;
    const float wz0 = 1.0f - wz1, wy0 = 1.0f - wy1;

    // post-clamp indices are guaranteed in [0, dim-1] -> unsigned is safe
    const uint32_t iz0 = (uint32_t)(int)fz;
    const uint32_t iy0 = (uint32_t)(int)fy;
    const uint32_t ix0 = (uint32_t)(int)fx;
    const uint32_t iz1 = min(iz0 + 1u, Dd - 1u);
    const uint32_t iy1 = min(iy0 + 1u, Hd - 1u);

    // x-corners are contiguous in memory: load both from base xb = min(ix0, W-2)
    // and fold the x==W-1 clamp into the x-weights (keeps the 24B load
    // in-bounds and unconditional).
    const uint32_t xb = min(ix0, Wd - 2u);
    const bool xclamped = (ix0 > xb);                 // ix0 == Wd-1
    const float ax_lo = xclamped ? 0.0f : (1.0f - wx1);
    const float ax_hi = xclamped ? 1.0f : wx1;        // wx0+wx1 == 1 when clamped

    // 32-bit BYTE offsets of the four (z,y) gather rows
    const uint32_t rz0 = (b * Dd + iz0) * Hd;
    const uint32_t rz1 = (b * Dd + iz1) * Hd;
    const uint32_t off00 = ((rz0 + iy0) * Wd + xb) * 12u;   // (z0,y0)
    const uint32_t off01 = ((rz0 + iy1) * Wd + xb) * 12u;   // (z0,y1)
    const uint32_t off10 = ((rz1 + iy0) * Wd + xb) * 12u;   // (z1,y0)
    const uint32_t off11 = ((rz1 + iy1) * Wd + xb) * 12u;   // (z1,y1)

    const float w00 = wz0 * wy0;
    const float w01 = wz0 * wy1;
    const float w10 = wz1 * wy0;
    const float w11 = wz1 * wy1;

    float acc0 = 0.0f, acc1 = 0.0f, acc2 = 0.0f;

    // Each row gather pulls 2 x-adjacent voxels = 6 contiguous floats (24B):
    // 4 gather loads/voxel instead of 8. d1 has heavy neighbor reuse and its
    // 165MB working set fits the 192MB L2 -> keep these loads temporal.
#define GATHER_ROW(OFF, WZY)                                             \
    {                                                                    \
        const float* __restrict__ p = byte_off(d1, (OFF));               \
        const float r0 = p[0], r1 = p[1], r2 = p[2];                     \
        const float r3 = p[3], r4 = p[4], r5 = p[5];                     \
        const float wl = (WZY) * ax_lo;                                  \
        const float wh = (WZY) * ax_hi;                                  \
        acc0 += wl * r0 + wh * r3;                                       \
        acc1 += wl * r1 + wh * r4;                                       \
        acc2 += wl * r2 + wh * r5;                                       \
    }

    GATHER_ROW(off00, w00)
    GATHER_ROW(off01, w01)
    GATHER_ROW(off10, w10)
    GATHER_ROW(off11, w11)
#undef GATHER_ROW

    // --- streamed output write (non-temporal) ---
    float* po = byte_off(out, voxB);
    __builtin_nontemporal_store(acc0 + d2z, po + 0);
    __builtin_nontemporal_store(acc1 + d2y, po + 1);
    __builtin_nontemporal_store(acc2 + d2x, po + 2);
}

extern "C" void kernel_launch(void* const* d_in, const int* in_sizes, int n_in,
                              void* d_out, int out_size, void* d_ws, size_t ws_size,
                              hipStream_t stream) {
    (void)in_sizes; (void)n_in; (void)out_size; (void)d_ws; (void)ws_size;

    const float* disp1 = (const float*)d_in[0];
    const float* disp2 = (const float*)d_in[1];
    float* out = (float*)d_out;

    dim3 block(Wd, 1, 1);        // 224 threads = 7 wave32
    dim3 grid(Hd, Dd, Bd);       // one block per (y-row, z, b)
    hipLaunchKernelGGL(compose_disp_kernel, grid, block, 0, stream,
                       disp1, disp2, out);
}